// CausalFFTConv_1348619731143
// MI455X (gfx1250) — compile-verified
//
#include <hip/hip_runtime.h>
#include <math.h>

// Causal conv with k[t,d] = exp(-|decay|t)cos(freq t) == Re(lambda^t),
// lambda = exp(-|decay|)*e^{i freq}, |lambda| <= 1. Exact reformulation as a
// first-order complex linear recurrence h[t] = lambda*h[t-1] + x[t], y=Re(h).
//
// 3-pass chunked scan. HBM traffic: x read (134MB, warms 192MB L2), y write
// (134MB, non-temporal so it doesn't evict x), pass-3 x re-read mostly L2
// hits -> ~11.5us at 23.3 TB/s. Compute (~0.2 GFLOP) is negligible; WMMA is
// inapplicable (per-channel kernel -> no shared matrix operand).
//
// CDNA5 paths: global_load_async_to_lds_b128 double-buffered streaming
// (ASYNCcnt) + s_wait_asynccnt; 512B per wave-instruction.

#define B_      4
#define T_      8192
#define D_      1024
#define LCHUNK  256            // chunk length along T
#define CCHUNK  (T_ / LCHUNK)  // 32 chunks
#define TILE    16             // rows per async LDS tile
#define NTILE   (LCHUNK / TILE)
#define VEC     4              // channels per thread (float4)
#define DQ      (D_ / VEC)     // 256 vector-channels
#define VBLOCK  128            // threads per block in vector kernels
#define PBLOCK  256
#define LOG2L   8              // LCHUNK == 2^8

typedef float v4f __attribute__((ext_vector_type(4)));  // native vector (16B)

// ---- CDNA5 async global->LDS copy (16 bytes/lane, 512B/wave) ----
__device__ __forceinline__ void async_copy_b128(const float* gptr, const v4f* lptr) {
    // vdst = 32-bit workgroup-relative LDS byte address (low 32 bits of the
    // flat shared pointer: LDS aperture is {hi32, 32'b0}).
    asm volatile("global_load_async_to_lds_b128 %0, %1, off"
                 :: "v"((unsigned)(unsigned long long)(lptr)), "v"(gptr)
                 : "memory");
}
#define ASYNC_WAIT(nstr) asm volatile("s_wait_asynccnt " nstr ::: "memory")

__device__ __forceinline__ void get_lambda(const float* __restrict__ decay,
                                           const float* __restrict__ freq,
                                           int d, float& lr, float& li) {
    float a  = fabsf(decay[d]);
    float er = expf(-a);
    float s, c;
    sincosf(freq[d], &s, &c);
    lr = er * c;
    li = er * s;
}

// ---------------- Pass 1: per-chunk local end state ----------------
// S_c = sum_{i=0}^{L-1} lambda^{L-1-i} x[cL+i]   (zero-init recurrence)
__global__ __launch_bounds__(VBLOCK) void chunk_state_kernel(
        const float* __restrict__ x,
        const float* __restrict__ decay,
        const float* __restrict__ freq,
        float* __restrict__ Sr, float* __restrict__ Si) {
    __shared__ v4f buf[2][TILE][VBLOCK];

    const int tid = threadIdx.x;
    const int gid = blockIdx.x * VBLOCK + tid;
    const int dq  = gid % DQ;
    const int bc  = gid / DQ;
    const int c   = bc % CCHUNK;
    const int b   = bc / CCHUNK;
    const int d0  = dq * VEC;

    const float* xp = x + ((long)(b * T_ + c * LCHUNK)) * D_ + d0;

    // prologue: stage tile 0
#pragma unroll
    for (int r = 0; r < TILE; ++r)
        async_copy_b128(xp + (long)r * D_, &buf[0][r][tid]);

    float lr[VEC], li[VEC], sr[VEC], si[VEC];
#pragma unroll
    for (int v = 0; v < VEC; ++v) {
        get_lambda(decay, freq, d0 + v, lr[v], li[v]);
        sr[v] = 0.f; si[v] = 0.f;
    }

    for (int j = 0; j < NTILE; ++j) {
        if (j + 1 < NTILE) {
            const float* xt = xp + (long)(j + 1) * TILE * D_;
#pragma unroll
            for (int r = 0; r < TILE; ++r)
                async_copy_b128(xt + (long)r * D_, &buf[(j + 1) & 1][r][tid]);
            ASYNC_WAIT("16");   // tile j complete (async loads finish in order)
        } else {
            ASYNC_WAIT("0");
        }
#pragma unroll
        for (int r = 0; r < TILE; ++r) {
            v4f xv = buf[j & 1][r][tid];
#pragma unroll
            for (int v = 0; v < VEC; ++v) {
                float nr = fmaf(lr[v], sr[v], fmaf(-li[v], si[v], xv[v]));
                float ni = fmaf(li[v], sr[v], lr[v] * si[v]);
                sr[v] = nr; si[v] = ni;
            }
        }
    }
    const int idx = (b * CCHUNK + c) * D_ + d0;
    v4f outr, outi;
#pragma unroll
    for (int v = 0; v < VEC; ++v) { outr[v] = sr[v]; outi[v] = si[v]; }
    *reinterpret_cast<v4f*>(Sr + idx) = outr;
    *reinterpret_cast<v4f*>(Si + idx) = outi;
}

// ---------------- Pass 2: prefix over chunk states ----------------
// carry_{c} = Lambda*carry_{c-1} + S_{c-1},  Lambda = lambda^LCHUNK.
// H_c = state just before chunk c (h[cL-1]).
__global__ __launch_bounds__(PBLOCK) void prefix_kernel(
        const float* __restrict__ decay,
        const float* __restrict__ freq,
        const float* __restrict__ Sr, const float* __restrict__ Si,
        float* __restrict__ Hr, float* __restrict__ Hi) {
    const int gid = blockIdx.x * PBLOCK + threadIdx.x;
    const int d   = gid % D_;
    const int b   = gid / D_;

    float lr, li;
    get_lambda(decay, freq, d, lr, li);

    // Lambda = lambda^(2^LOG2L) by repeated complex squaring
    float Lr = lr, Li = li;
#pragma unroll
    for (int k = 0; k < LOG2L; ++k) {
        float nr = Lr * Lr - Li * Li;
        float ni = 2.f * Lr * Li;
        Lr = nr; Li = ni;
    }

    float cr = 0.f, ci = 0.f;
#pragma unroll
    for (int c = 0; c < CCHUNK; ++c) {
        const int idx = (b * CCHUNK + c) * D_ + d;
        Hr[idx] = cr;
        Hi[idx] = ci;
        float sr = Sr[idx], si = Si[idx];
        float nr = fmaf(Lr, cr, fmaf(-Li, ci, sr));
        float ni = fmaf(Li, cr, fmaf(Lr, ci, si));
        cr = nr; ci = ni;
    }
}

// ---------------- Pass 3: seeded replay, emit y ----------------
__global__ __launch_bounds__(VBLOCK) void apply_kernel(
        const float* __restrict__ x,
        const float* __restrict__ decay,
        const float* __restrict__ freq,
        const float* __restrict__ Hr, const float* __restrict__ Hi,
        float* __restrict__ y) {
    __shared__ v4f buf[2][TILE][VBLOCK];

    const int tid = threadIdx.x;
    const int gid = blockIdx.x * VBLOCK + tid;
    const int dq  = gid % DQ;
    const int bc  = gid / DQ;
    const int c   = bc % CCHUNK;
    const int b   = bc / CCHUNK;
    const int d0  = dq * VEC;

    const long base = ((long)(b * T_ + c * LCHUNK)) * D_ + d0;
    const float* xp = x + base;
    float*       yp = y + base;

#pragma unroll
    for (int r = 0; r < TILE; ++r)
        async_copy_b128(xp + (long)r * D_, &buf[0][r][tid]);

    float lr[VEC], li[VEC], hr[VEC], hi[VEC];
    const int idx = (b * CCHUNK + c) * D_ + d0;
    v4f h0r = *reinterpret_cast<const v4f*>(Hr + idx);
    v4f h0i = *reinterpret_cast<const v4f*>(Hi + idx);
#pragma unroll
    for (int v = 0; v < VEC; ++v) {
        get_lambda(decay, freq, d0 + v, lr[v], li[v]);
        hr[v] = h0r[v]; hi[v] = h0i[v];
    }

    for (int j = 0; j < NTILE; ++j) {
        if (j + 1 < NTILE) {
            const float* xt = xp + (long)(j + 1) * TILE * D_;
#pragma unroll
            for (int r = 0; r < TILE; ++r)
                async_copy_b128(xt + (long)r * D_, &buf[(j + 1) & 1][r][tid]);
            ASYNC_WAIT("16");
        } else {
            ASYNC_WAIT("0");
        }
#pragma unroll
        for (int r = 0; r < TILE; ++r) {
            v4f xv = buf[j & 1][r][tid];
            v4f outv;
#pragma unroll
            for (int v = 0; v < VEC; ++v) {
                float nr = fmaf(lr[v], hr[v], fmaf(-li[v], hi[v], xv[v]));
                float ni = fmaf(li[v], hr[v], lr[v] * hi[v]);
                hr[v] = nr; hi[v] = ni;
                outv[v] = nr;
            }
            // y = Re(h); non-temporal so the output stream doesn't evict x
            // from the 192MB L2 (x re-read in this pass should hit L2).
            __builtin_nontemporal_store(
                outv, reinterpret_cast<v4f*>(yp + (long)(j * TILE + r) * D_));
        }
    }
}

extern "C" void kernel_launch(void* const* d_in, const int* in_sizes, int n_in,
                              void* d_out, int out_size, void* d_ws, size_t ws_size,
                              hipStream_t stream) {
    const float* x     = (const float*)d_in[0];
    const float* decay = (const float*)d_in[1];
    const float* freq  = (const float*)d_in[2];
    float*       y     = (float*)d_out;

    const int N = B_ * CCHUNK * D_;          // 131072 chunk states
    float* Sr = (float*)d_ws;                // 4 planes -> 2 MB of d_ws
    float* Si = Sr + N;
    float* Hr = Si + N;
    float* Hi = Hr + N;

    chunk_state_kernel<<<(B_ * CCHUNK * DQ) / VBLOCK, dim3(VBLOCK), 0, stream>>>(
        x, decay, freq, Sr, Si);
    prefix_kernel<<<(B_ * D_) / PBLOCK, dim3(PBLOCK), 0, stream>>>(
        decay, freq, Sr, Si, Hr, Hi);
    apply_kernel<<<(B_ * CCHUNK * DQ) / VBLOCK, dim3(VBLOCK), 0, stream>>>(
        x, decay, freq, Hr, Hi, y);
}